// LoRACLayer_47562467836478
// MI455X (gfx1250) — compile-verified
//
#include <hip/hip_runtime.h>

typedef float v2f __attribute__((ext_vector_type(2)));
typedef float v4f __attribute__((ext_vector_type(4)));
typedef float v8f __attribute__((ext_vector_type(8)));

#define BN   32
#define COUT 64
#define HH   224
#define WW   224
#define HW   (HH * WW)      // 50176
#define SCALING 2.0f

// LDS layout (float offsets)
#define XS_OFF   0          // 12 x 20 halo of x                (240)
#define LA_OFF   240        // lora_A                           (144)
#define XDS_OFF  384        // x_down tile: 16 r x 10 x 20      (3200)
#define BS2_OFF  3584       // packed B: 80 k-pairs x 160 flts  (12800)
#define SMEM_FLOATS 16384   // 64 KB

// Packed B-matrix entry. K ordering: k = t*16 + r for the LoRA part
// (t = tap 0..8, r = lora channel 0..15), k = 144+t for the main conv taps,
// k = 153..159 zero padding. LoRA scaling folded in here.
__device__ __forceinline__ float bval(int k, int c,
                                      const float* __restrict__ weight,
                                      const float* __restrict__ lora_B) {
  if (k < 144) {
    int t = k >> 4, r = k & 15;
    int kh = t / 3, kw = t - 3 * kh;
    // lora_B layout: (COUT, KH, KW, R)
    return SCALING * lora_B[((c * 3 + kh) * 3 + kw) * 16 + r];
  } else if (k < 153) {
    int t = k - 144;
    return weight[c * 9 + t];   // weight layout: (COUT, 1, 3, 3)
  }
  return 0.0f;
}

__global__ __launch_bounds__(256) void lora_conv_wmma(
    const float* __restrict__ x, const float* __restrict__ weight,
    const float* __restrict__ bias, const float* __restrict__ lora_A,
    const float* __restrict__ lora_B, float* __restrict__ out) {
  __shared__ __align__(16) float smem[SMEM_FLOATS];

  const int tid   = threadIdx.x;
  const int lane  = tid & 31;
  const int wave  = tid >> 5;
  const int n     = lane & 15;   // N index within a 16-wide tile / M pixel index
  const int khalf = lane >> 4;   // selects K pair {0,1} vs {2,3} of each wmma

  int bid = blockIdx.x;
  const int wb = bid % 14; bid /= 14;
  const int hb = bid % 28;
  const int bb = bid / 28;
  const int w0 = wb * 16;
  const int h0 = hb * 8;

  // ---- Phase 1: x halo (rows h0-2..h0+9, cols w0-2..w0+17) + lora_A to LDS
  if (tid < 240) {
    int row = tid / 20, col = tid % 20;
    int gh = h0 - 2 + row, gw = w0 - 2 + col;
    float v = 0.0f;
    if (gh >= 0 && gh < HH && gw >= 0 && gw < WW)
      v = x[(size_t)bb * HW + gh * WW + gw];
    smem[XS_OFF + tid] = v;
  }
  if (tid < 144) smem[LA_OFF + tid] = lora_A[tid];

  // ---- Phase 2: packed B matrix, rows paired as float2 for ds_load_b64.
  // Row-pair p stride = 160 floats (640B) so lane-halves hit disjoint banks.
  for (int e = tid; e < 5120; e += 256) {
    int p = e >> 6, c = e & 63;
    smem[BS2_OFF + p * 160 + 2 * c + 0] = bval(2 * p + 0, c, weight, lora_B);
    smem[BS2_OFF + p * 160 + 2 * c + 1] = bval(2 * p + 1, c, weight, lora_B);
  }
  __syncthreads();

  // ---- Phase 3: x_down tile = conv3x3(x, lora_A), 16 r x 10 rows x 18 cols
  for (int e = tid; e < 2880; e += 256) {
    int r   = e / 180;
    int rem = e - r * 180;
    int row = rem / 18;
    int col = rem - row * 18;
    float acc = 0.0f;
#pragma unroll
    for (int kh = 0; kh < 3; ++kh)
#pragma unroll
      for (int kw = 0; kw < 3; ++kw)
        acc += smem[LA_OFF + r * 9 + kh * 3 + kw] *
               smem[XS_OFF + (row + kh) * 20 + (col + kw)];
    smem[XDS_OFF + r * 200 + row * 20 + col] = acc;
  }
  __syncthreads();

  // ---- Phase 4: implicit GEMM. Per wave: M = 16 pixels of row h0+wave,
  // N = 64 (4 tiles), K = 160 -> 39 useful k-steps of V_WMMA_F32_16X16X4_F32.
  const int aoff  = wave * 20 + n;               // lane-dependent (row, pixel) part
  const int bbase = BS2_OFF + khalf * 160 + 2 * n;
  v8f acc0 = {0, 0, 0, 0, 0, 0, 0, 0};
  v8f acc1 = acc0, acc2 = acc0, acc3 = acc0;

#pragma unroll
  for (int s = 0; s < 39; ++s) {
    v2f a;
    if (s < 36) {
      // LoRA region: k = t*16 + r; A[m][k] = x_down[r][wave+kh][n+kw]
      int k0 = 4 * s + 2 * khalf;
      int t0 = k0 >> 4;
      int r0 = k0 & 15;                 // k0 even -> r0 <= 14, pair shares t0
      int kh = (t0 * 11) >> 5;          // t0/3 for t0 < 9
      int ao = XDS_OFF + r0 * 200 + t0 + 17 * kh + aoff; // +kh*20+kw folded
      a.x = smem[ao];
      a.y = smem[ao + 200];
    } else if (s < 38) {
      // main-conv taps: k = 144 + t; A[m][k] = x[wave+kh+1][n+kw+1]
      int t0  = (s - 36) * 4 + 2 * khalf;
      int kh0 = (t0 * 11) >> 5;
      int kh1 = ((t0 + 1) * 11) >> 5;
      a.x = smem[XS_OFF + 21 + t0 + 17 * kh0 + aoff];
      a.y = smem[XS_OFF + 21 + (t0 + 1) + 17 * kh1 + aoff];
    } else {
      // s == 38: k = 152 (tap t=8) valid for khalf==0 only; rest is zero pad
      float v = smem[XS_OFF + 21 + 8 + 17 * 2 + aoff];
      a.x = khalf ? 0.0f : v;
      a.y = 0.0f;
    }
    const float* bp = smem + bbase + s * 320;
    v2f b0 = *(const v2f*)(bp + 0);
    v2f b1 = *(const v2f*)(bp + 32);
    v2f b2 = *(const v2f*)(bp + 64);
    v2f b3 = *(const v2f*)(bp + 96);
    acc0 = __builtin_amdgcn_wmma_f32_16x16x4_f32(false, a, false, b0, (short)0, acc0, false, false);
    acc1 = __builtin_amdgcn_wmma_f32_16x16x4_f32(false, a, false, b1, (short)0, acc1, false, false);
    acc2 = __builtin_amdgcn_wmma_f32_16x16x4_f32(false, a, false, b2, (short)0, acc2, false, false);
    acc3 = __builtin_amdgcn_wmma_f32_16x16x4_f32(false, a, false, b3, (short)0, acc3, false, false);
  }
  __syncthreads();   // everyone done reading Bs2 -> reuse as store staging

  // ---- Phase 5: stage (acc + bias) in LDS, then coalesced b128 stores.
  // C/D layout: lane n = N, half = khalf; vgpr j holds M = j + 8*khalf.
  const int bufb = BS2_OFF + wave * 1024;     // buf[c][m], 64 x 16 floats
  {
    float bv0 = bias[n], bv1 = bias[16 + n], bv2 = bias[32 + n], bv3 = bias[48 + n];
    int base = bufb + n * 16 + 8 * khalf;
#define STAGE(ACC, NT, BV)                                                  \
    { v4f lo = {ACC[0] + BV, ACC[1] + BV, ACC[2] + BV, ACC[3] + BV};        \
      v4f hi = {ACC[4] + BV, ACC[5] + BV, ACC[6] + BV, ACC[7] + BV};        \
      *(v4f*)(smem + base + (NT) * 256) = lo;                               \
      *(v4f*)(smem + base + (NT) * 256 + 4) = hi; }
    STAGE(acc0, 0, bv0)
    STAGE(acc1, 1, bv1)
    STAGE(acc2, 2, bv2)
    STAGE(acc3, 3, bv3)
#undef STAGE
  }
  const int h = h0 + wave;
  const size_t obase = (size_t)bb * COUT * HW + (size_t)h * WW + w0;
#pragma unroll
  for (int it = 0; it < 8; ++it) {
    int idx = it * 32 + lane;       // 0..255 over (c, m/4)
    int c   = idx >> 2;
    int mq  = (idx & 3) << 2;
    v4f v = *(const v4f*)(smem + bufb + c * 16 + mq);
    *(v4f*)(out + obase + (size_t)c * HW + mq) = v;
  }
}

extern "C" void kernel_launch(void* const* d_in, const int* in_sizes, int n_in,
                              void* d_out, int out_size, void* d_ws, size_t ws_size,
                              hipStream_t stream) {
  (void)in_sizes; (void)n_in; (void)out_size; (void)d_ws; (void)ws_size;
  const float* x      = (const float*)d_in[0];
  const float* weight = (const float*)d_in[1];
  const float* bias   = (const float*)d_in[2];
  const float* lora_A = (const float*)d_in[3];
  const float* lora_B = (const float*)d_in[4];
  float* out = (float*)d_out;

  dim3 grid(BN * (HH / 8) * (WW / 16));   // 32 * 28 * 14 = 12544
  dim3 block(256);                         // 8 wave32
  hipLaunchKernelGGL(lora_conv_wmma, grid, block, 0, stream,
                     x, weight, bias, lora_A, lora_B, out);
}